// ResourceGraphEncoder_16690242912872
// MI455X (gfx1250) — compile-verified
//
#include <hip/hip_runtime.h>
#include <hip/hip_bf16.h>

typedef float v2f __attribute__((ext_vector_type(2)));
typedef float v8f __attribute__((ext_vector_type(8)));

// Hardware fp32 global atomic add (global_atomic_add_f32), not a CAS loop.
__device__ __forceinline__ void atomAddF(float* p, float v) {
    unsafeAtomicAdd(p, v);
}

// ---------------------------------------------------------------------------
// Kernel 1: per-edge degree count + layer-1 feature scatter (2 features)
// ---------------------------------------------------------------------------
__global__ void deg_agg1_kernel(const float* __restrict__ x,
                                const int* __restrict__ src,
                                const int* __restrict__ dst,
                                float* __restrict__ cnt,
                                float* __restrict__ agg1,
                                int n_edges) {
    int e = blockIdx.x * blockDim.x + threadIdx.x;
    if (e >= n_edges) return;
    int s = src[e];
    int d = dst[e];
    float2 xv = *(const float2*)(x + (size_t)s * 2);
    atomAddF(&cnt[d], 1.0f);
    atomAddF(&agg1[(size_t)d * 2 + 0], xv.x);
    atomAddF(&agg1[(size_t)d * 2 + 1], xv.y);
}

// ---------------------------------------------------------------------------
// Kernel 2: layer-1 node update: h1 = mean @ W1l + b1l + x @ W1r   (K=2: scalar)
// ---------------------------------------------------------------------------
__global__ void l1node_kernel(const float* __restrict__ x,
                              const float* __restrict__ agg1,
                              const float* __restrict__ cnt,
                              const float* __restrict__ W1l,
                              const float* __restrict__ b1l,
                              const float* __restrict__ W1r,
                              float* __restrict__ h1,
                              int n_nodes) {
    int i = blockIdx.x * blockDim.x + threadIdx.x;
    if (i >= n_nodes * 64) return;
    int node = i >> 6;
    int f = i & 63;
    float c = cnt[node];
    float inv = 1.0f / (c > 1.0f ? c : 1.0f);
    float m0 = agg1[(size_t)node * 2 + 0] * inv;
    float m1 = agg1[(size_t)node * 2 + 1] * inv;
    float x0 = x[(size_t)node * 2 + 0];
    float x1 = x[(size_t)node * 2 + 1];
    h1[i] = m0 * W1l[f] + m1 * W1l[64 + f] + b1l[f]
          + x0 * W1r[f] + x1 * W1r[64 + f];
}

// ---------------------------------------------------------------------------
// Kernel 3: column sums / sums-of-squares for BatchNorm.
// stride is a multiple of 64 -> each thread's column (i&63) is fixed.
// ---------------------------------------------------------------------------
__global__ void colstats_kernel(const float* __restrict__ h,
                                float* __restrict__ S1,
                                float* __restrict__ S2,
                                int total) {
    __shared__ float l1[64];
    __shared__ float l2[64];
    if (threadIdx.x < 64) { l1[threadIdx.x] = 0.0f; l2[threadIdx.x] = 0.0f; }
    __syncthreads();
    int stride = gridDim.x * blockDim.x;
    float s1 = 0.0f, s2 = 0.0f;
    for (int i = blockIdx.x * blockDim.x + threadIdx.x; i < total; i += stride) {
        float v = h[i];
        s1 += v;
        s2 += v * v;
    }
    int f = threadIdx.x & 63;
    atomicAdd(&l1[f], s1);
    atomicAdd(&l2[f], s2);
    __syncthreads();
    if (threadIdx.x < 64) {
        atomicAdd(&S1[threadIdx.x], l1[threadIdx.x]);
        atomicAdd(&S2[threadIdx.x], l2[threadIdx.x]);
    }
}

// ---------------------------------------------------------------------------
// Kernel 4: BN (training-mode) + ReLU, in place
// ---------------------------------------------------------------------------
__global__ void bnrelu_kernel(float* __restrict__ h,
                              const float* __restrict__ S1,
                              const float* __restrict__ S2,
                              const float* __restrict__ gamma,
                              const float* __restrict__ beta,
                              int n_nodes) {
    int i = blockIdx.x * blockDim.x + threadIdx.x;
    if (i >= n_nodes * 64) return;
    int f = i & 63;
    float invN = 1.0f / (float)n_nodes;
    float m = S1[f] * invN;
    float v = S2[f] * invN - m * m;
    float sc = rsqrtf(v + 1e-5f) * gamma[f];
    float val = (h[i] - m) * sc + beta[f];
    h[i] = val > 0.0f ? val : 0.0f;
}

// ---------------------------------------------------------------------------
// Kernel 5: heavy layer-2 scatter: 16 threads per edge, float4 gather + 4
// hardware fp32 atomics each. Consecutive 16 threads read one coalesced
// 256-byte h1 row.
// ---------------------------------------------------------------------------
__global__ void agg2_kernel(const float* __restrict__ h1,
                            const int* __restrict__ src,
                            const int* __restrict__ dst,
                            float* __restrict__ agg2,
                            int n_edges) {
    int t = blockIdx.x * blockDim.x + threadIdx.x;
    int e = t >> 4;
    if (e >= n_edges) return;
    int c4 = t & 15;
    int s = src[e];
    int d = dst[e];
    float4 v = *(const float4*)(h1 + (size_t)s * 64 + c4 * 4);
    float* a = agg2 + (size_t)d * 64 + c4 * 4;
    atomAddF(a + 0, v.x);
    atomAddF(a + 1, v.y);
    atomAddF(a + 2, v.z);
    atomAddF(a + 3, v.w);
}

// ---------------------------------------------------------------------------
// Kernel 6: layer-2 node update with V_WMMA_F32_16X16X4_F32.
// One wave per 16x16 output tile; both GEMMs (mean2@W2l and h1@W2r) fused
// into one accumulator initialized with the bias.
// A layout (16x4 f32, ISA 7.12.2): lanes 0-15 hold M=lane K={0,1} in
// vgpr{0,1}; lanes 16-31 hold K={2,3}. B (4x16): row K striped across lanes
// within a vgpr, mirrored split. D: vgpr r = rows r (lanes 0-15) / r+8
// (lanes 16-31), N = lane&15.
// ---------------------------------------------------------------------------
__global__ void __launch_bounds__(256)
l2node_wmma_kernel(const float* __restrict__ agg2,
                   const float* __restrict__ cnt,
                   const float* __restrict__ h1,
                   const float* __restrict__ W2l,
                   const float* __restrict__ b2l,
                   const float* __restrict__ W2r,
                   float* __restrict__ h2,
                   int n_nodes) {
    int wave = blockIdx.x * (blockDim.x >> 5) + (threadIdx.x >> 5);
    int lane = threadIdx.x & 31;
    int tile_m = wave >> 2;      // N/16 row tiles
    int tile_n = wave & 3;       // 64/16 col tiles
    int m0 = tile_m * 16;
    int n0 = tile_n * 16;
    int lm = lane & 15;
    int khalf = (lane >> 4) * 2;

    int arow = m0 + lm;
    if (arow >= n_nodes) arow = n_nodes - 1;   // clamp loads (stores guarded)
    float c = cnt[arow];
    float inv = 1.0f / (c > 1.0f ? c : 1.0f);
    const float* aggRow = agg2 + (size_t)arow * 64;
    const float* h1Row  = h1   + (size_t)arow * 64;

    float bval = b2l[n0 + lm];
    v8f acc;
#pragma unroll
    for (int r = 0; r < 8; ++r) acc[r] = bval;

#pragma unroll
    for (int kk = 0; kk < 16; ++kk) {
        int k = kk * 4 + khalf;
        v2f a1, b1v, a2, b2v;
        a1.x = aggRow[k] * inv;
        a1.y = aggRow[k + 1] * inv;
        b1v.x = W2l[k * 64 + n0 + lm];
        b1v.y = W2l[(k + 1) * 64 + n0 + lm];
        acc = __builtin_amdgcn_wmma_f32_16x16x4_f32(
            false, a1, false, b1v, (short)0, acc, false, false);
        a2.x = h1Row[k];
        a2.y = h1Row[k + 1];
        b2v.x = W2r[k * 64 + n0 + lm];
        b2v.y = W2r[(k + 1) * 64 + n0 + lm];
        acc = __builtin_amdgcn_wmma_f32_16x16x4_f32(
            false, a2, false, b2v, (short)0, acc, false, false);
    }

    int col = n0 + lm;
    int rbase = m0 + (lane >> 4) * 8;
#pragma unroll
    for (int r = 0; r < 8; ++r) {
        int row = rbase + r;
        if (row < n_nodes) h2[(size_t)row * 64 + col] = acc[r];
    }
}

// ---------------------------------------------------------------------------
// Kernel 7: BN + ReLU + column max (uint-bit atomicMax; post-ReLU values >= 0
// so non-negative float bits compare correctly as uint; out is zeroed).
// ---------------------------------------------------------------------------
__global__ void bnrelumax_kernel(const float* __restrict__ h,
                                 const float* __restrict__ S1,
                                 const float* __restrict__ S2,
                                 const float* __restrict__ gamma,
                                 const float* __restrict__ beta,
                                 float* __restrict__ out,
                                 int n_nodes) {
    __shared__ unsigned lmax[64];
    if (threadIdx.x < 64) lmax[threadIdx.x] = 0u;
    __syncthreads();
    int total = n_nodes * 64;
    int stride = gridDim.x * blockDim.x;
    int f = threadIdx.x & 63;
    float invN = 1.0f / (float)n_nodes;
    float m = S1[f] * invN;
    float v = S2[f] * invN - m * m;
    float sc = rsqrtf(v + 1e-5f) * gamma[f];
    float sh = beta[f] - m * sc;
    float best = 0.0f;   // ReLU folded into the max (init 0)
    for (int i = blockIdx.x * blockDim.x + threadIdx.x; i < total; i += stride) {
        float val = h[i] * sc + sh;
        best = fmaxf(best, val);
    }
    atomicMax(&lmax[f], __float_as_uint(best));
    __syncthreads();
    if (threadIdx.x < 64)
        atomicMax((unsigned*)&out[threadIdx.x], lmax[threadIdx.x]);
}

// ---------------------------------------------------------------------------
// Host launcher
// ---------------------------------------------------------------------------
extern "C" void kernel_launch(void* const* d_in, const int* in_sizes, int n_in,
                              void* d_out, int out_size, void* d_ws, size_t ws_size,
                              hipStream_t stream) {
    const float* x   = (const float*)d_in[0];
    const int*   ei  = (const int*)d_in[1];
    const float* W1l = (const float*)d_in[2];
    const float* b1l = (const float*)d_in[3];
    const float* W1r = (const float*)d_in[4];
    const float* W2l = (const float*)d_in[5];
    const float* b2l = (const float*)d_in[6];
    const float* W2r = (const float*)d_in[7];
    const float* g1  = (const float*)d_in[8];
    const float* be1 = (const float*)d_in[9];
    const float* g2  = (const float*)d_in[10];
    const float* be2 = (const float*)d_in[11];
    float* out = (float*)d_out;

    const int N = in_sizes[0] / 2;      // 100000
    const int E = in_sizes[1] / 2;      // 1600000
    const int* src = ei;
    const int* dst = ei + E;

    // Workspace layout (floats)
    float* ws    = (float*)d_ws;
    float* cnt   = ws;                       // N
    float* agg1  = cnt + (size_t)N;          // 2N
    float* h1    = agg1 + (size_t)2 * N;     // 64N
    float* agg2  = h1 + (size_t)64 * N;      // 64N
    float* h2    = agg2 + (size_t)64 * N;    // 64N
    float* stats = h2 + (size_t)64 * N;      // 256  (S1,S2,S3,S4)
    float* S1 = stats, *S2 = stats + 64, *S3 = stats + 128, *S4 = stats + 192;

    // Zero accumulators + output (every call; capture-safe)
    hipMemsetAsync(cnt,  0, (size_t)3 * N * sizeof(float), stream);   // cnt+agg1
    hipMemsetAsync(agg2, 0, (size_t)64 * N * sizeof(float), stream);
    hipMemsetAsync(stats, 0, 256 * sizeof(float), stream);
    hipMemsetAsync(out,  0, 64 * sizeof(float), stream);

    const int B = 256;

    // 1) degree + layer-1 scatter
    deg_agg1_kernel<<<(E + B - 1) / B, B, 0, stream>>>(x, src, dst, cnt, agg1, E);

    // 2) layer-1 node update
    l1node_kernel<<<((size_t)N * 64 + B - 1) / B, B, 0, stream>>>(
        x, agg1, cnt, W1l, b1l, W1r, h1, N);

    // 3) BN1 stats, apply BN1+ReLU in place
    colstats_kernel<<<512, B, 0, stream>>>(h1, S1, S2, N * 64);
    bnrelu_kernel<<<((size_t)N * 64 + B - 1) / B, B, 0, stream>>>(
        h1, S1, S2, g1, be1, N);

    // 4) heavy layer-2 scatter: 16 threads/edge
    agg2_kernel<<<(((size_t)E * 16) + B - 1) / B, B, 0, stream>>>(
        h1, src, dst, agg2, E);

    // 5) layer-2 node update via WMMA (one wave per 16x16 tile)
    int tilesM = (N + 15) / 16;
    int waves = tilesM * 4;
    int wavesPerBlock = B / 32;
    l2node_wmma_kernel<<<(waves + wavesPerBlock - 1) / wavesPerBlock, B, 0, stream>>>(
        agg2, cnt, h1, W2l, b2l, W2r, h2, N);

    // 6) BN2 stats + BN2+ReLU+column-max
    colstats_kernel<<<512, B, 0, stream>>>(h2, S3, S4, N * 64);
    bnrelumax_kernel<<<512, B, 0, stream>>>(h2, S3, S4, g2, be2, out, N);
}